// ContrastLoss_15255723836120
// MI455X (gfx1250) — compile-verified
//
#include <hip/hip_runtime.h>
#include <math.h>

#define BATCH 32768
#define NCLS  1000
#define FDIM  512
#define NTILES 63            // ceil(1000/16)
#define CPAD  (NTILES * 16)  // 1008
#define INV_TEMP 10.0f
#define MOM 0.9f

typedef float v2f __attribute__((ext_vector_type(2)));
typedef float v8f __attribute__((ext_vector_type(8)));

__device__ __forceinline__ float atomicAddF32(float* p, float v) {
  return __hip_atomic_fetch_add(p, v, __ATOMIC_RELAXED, __HIP_MEMORY_SCOPE_AGENT);
}

// ---------------- Kernel 1: per-row max, sumexp (CE) + sumexp (T), CE accum --------
__global__ void k1_row_stats(const float* __restrict__ logits,
                             const int* __restrict__ labels,
                             float* __restrict__ rowmax,
                             float* __restrict__ sumexp_t,
                             float* __restrict__ ce_acc) {
  __shared__ float sred[256];
  const int row = blockIdx.x;
  const int t = threadIdx.x;
  const float* x = logits + (size_t)row * NCLS;
  float v[4];
#pragma unroll
  for (int i = 0; i < 4; ++i) {
    int c = t + i * 256;
    v[i] = (c < NCLS) ? x[c] : -__builtin_inff();
  }
  float m = fmaxf(fmaxf(v[0], v[1]), fmaxf(v[2], v[3]));
  sred[t] = m; __syncthreads();
  for (int o = 128; o > 0; o >>= 1) { if (t < o) sred[t] = fmaxf(sred[t], sred[t + o]); __syncthreads(); }
  m = sred[0]; __syncthreads();

  float s_ce = 0.f, s_t = 0.f;
#pragma unroll
  for (int i = 0; i < 4; ++i) {
    int c = t + i * 256;
    if (c < NCLS) { float d = v[i] - m; s_ce += __expf(d); s_t += __expf(d * INV_TEMP); }
  }
  sred[t] = s_ce; __syncthreads();
  for (int o = 128; o > 0; o >>= 1) { if (t < o) sred[t] += sred[t + o]; __syncthreads(); }
  float sum_ce = sred[0]; __syncthreads();
  sred[t] = s_t; __syncthreads();
  for (int o = 128; o > 0; o >>= 1) { if (t < o) sred[t] += sred[t + o]; __syncthreads(); }
  float sum_t = sred[0];

  if (t == 0) {
    rowmax[row]   = m;
    sumexp_t[row] = sum_t;
    float logZ = m + __logf(sum_ce);
    float xlab = x[labels[row]];
    atomicAddF32(ce_acc, logZ - xlab);   // -logp[label]
  }
}

// ---------------- Kernel 2: segment-sum of features by label (L2-resident atomics) --
__global__ void k2_segsum(const float* __restrict__ features,
                          const int* __restrict__ labels,
                          float* __restrict__ seg_sum,
                          float* __restrict__ counts) {
  const int row = blockIdx.x;
  const int t = threadIdx.x;
  const int lab = labels[row];
  const float* f = features + (size_t)row * FDIM;
  float* dst = seg_sum + (size_t)lab * FDIM;
  atomicAddF32(&dst[t],        f[t]);
  atomicAddF32(&dst[t + 256],  f[t + 256]);
  if (t == 0) atomicAddF32(&counts[lab], 1.0f);
}

// ---------------- Kernel 3: momentum update + L2-normalize centers ------------------
__global__ void k3_centers(const float* __restrict__ centers,
                           const float* __restrict__ seg_sum,
                           const float* __restrict__ counts,
                           float* __restrict__ centers_n) {
  __shared__ float sred[256];
  const int c = blockIdx.x;
  const int t = threadIdx.x;
  float cnt = counts[c];
  float inv = 1.0f / fmaxf(cnt, 1.0f);
  bool present = cnt > 0.0f;
  const float* cc = centers + (size_t)c * FDIM;
  const float* ss = seg_sum + (size_t)c * FDIM;
  float u0 = present ? (MOM * cc[t]       + (1.0f - MOM) * ss[t]       * inv) : cc[t];
  float u1 = present ? (MOM * cc[t + 256] + (1.0f - MOM) * ss[t + 256] * inv) : cc[t + 256];
  sred[t] = u0 * u0 + u1 * u1; __syncthreads();
  for (int o = 128; o > 0; o >>= 1) { if (t < o) sred[t] += sred[t + o]; __syncthreads(); }
  float d = 1.0f / fmaxf(sqrtf(sred[0]), 1e-12f);
  float* out = centers_n + (size_t)c * FDIM;
  out[t]       = u0 * d;
  out[t + 256] = u1 * d;
}

// ---------------- Kernel 4: sim = N*N^T via V_WMMA_F32_16X16X4_F32 ------------------
// A 16x4 f32: lane l -> M = l&15, K-pair (l>>4)*2  => contiguous float2 per lane.
// B 4x16 f32: lane l -> N = l&15, same K-pair      => identical load pattern.
// C/D 16x16:  VGPR r -> M = r + ((l>>4)<<3), N = l&15.
__global__ void k4_sim_wmma(const float* __restrict__ Nn,
                            float* __restrict__ sim) {
  const int lane  = threadIdx.x;                       // 0..31 (wave32)
  const int tileN = blockIdx.x;                        // 0..62
  const int tileM = blockIdx.y * blockDim.y + threadIdx.y;
  if (tileM >= NTILES) return;                         // whole wave exits: EXEC stays all-1s
  const int ml = lane & 15;
  const int kk = (lane >> 4) << 1;
  const float* arow = Nn + (size_t)(tileM * 16 + ml) * FDIM + kk;
  const float* brow = Nn + (size_t)(tileN * 16 + ml) * FDIM + kk;
  v8f acc = {0.f, 0.f, 0.f, 0.f, 0.f, 0.f, 0.f, 0.f};
#pragma unroll 4
  for (int k = 0; k < FDIM; k += 4) {
    v2f a = *(const v2f*)(arow + k);
    v2f b = *(const v2f*)(brow + k);
    acc = __builtin_amdgcn_wmma_f32_16x16x4_f32(false, a, false, b,
                                                (short)0, acc, false, false);
  }
  const int col   = tileN * 16 + ml;
  const int rbase = tileM * 16 + ((lane >> 4) << 3);
#pragma unroll
  for (int r = 0; r < 8; ++r)
    sim[(size_t)(rbase + r) * CPAD + col] = (acc[r] + 1.0f) * 0.5f;  // map to [0,1]
}

// ---------------- Kernel 5: contrast term (2nd logits pass, sim gather in L2) -------
__global__ void k5_contrast(const float* __restrict__ logits,
                            const int* __restrict__ labels,
                            const float* __restrict__ rowmax,
                            const float* __restrict__ sumexp_t,
                            const float* __restrict__ sim,
                            float* __restrict__ con_acc) {
  __shared__ float sred[256];
  const int row = blockIdx.x;
  const int t = threadIdx.x;
  const float m    = rowmax[row];
  const float invs = 1.0f / sumexp_t[row];
  const int lab    = labels[row];
  const float* x    = logits + (size_t)row * NCLS;
  const float* srow = sim + (size_t)lab * CPAD;
  float acc = 0.f;
#pragma unroll
  for (int i = 0; i < 4; ++i) {
    int c = t + i * 256;
    if (c < NCLS) {
      float p = __expf((x[c] - m) * INV_TEMP) * invs;   // softmax(logits/T)
      acc += __logf(1.0f - p * srow[c] + 1e-6f);
    }
  }
  sred[t] = acc; __syncthreads();
  for (int o = 128; o > 0; o >>= 1) { if (t < o) sred[t] += sred[t + o]; __syncthreads(); }
  if (t == 0) atomicAddF32(con_acc, sred[0]);
}

// ---------------- Kernel 6: finalize ------------------------------------------------
__global__ void k6_final(const float* __restrict__ accum, float* __restrict__ out) {
  float ce  = accum[0] / (float)BATCH;
  float con = -accum[1] / ((float)BATCH * (float)NCLS);
  out[0] = ce + 0.1f * con;
}

extern "C" void kernel_launch(void* const* d_in, const int* in_sizes, int n_in,
                              void* d_out, int out_size, void* d_ws, size_t ws_size,
                              hipStream_t stream) {
  const float* logits   = (const float*)d_in[0];
  const float* features = (const float*)d_in[1];
  const int*   labels   = (const int*)d_in[2];
  const float* centers  = (const float*)d_in[3];
  float* out = (float*)d_out;
  float* ws  = (float*)d_ws;

  // workspace layout (floats)
  float* rowmax    = ws;                         // 32768
  float* sumexp_t  = rowmax + BATCH;             // 32768
  float* accum     = sumexp_t + BATCH;           // 16  ([0]=ce, [1]=contrast)
  float* counts    = accum + 16;                 // 1024
  float* seg_sum   = counts + 1024;              // 1000*512
  float* centers_n = seg_sum + NCLS * FDIM;      // 1008*512 (pad rows stay zero)
  float* sim       = centers_n + CPAD * FDIM;    // 1008*1008

  // zero accumulators + counts + seg_sum + padded normalized centers (one contiguous span)
  size_t zero_bytes = (size_t)(16 + 1024 + NCLS * FDIM + CPAD * FDIM) * sizeof(float);
  hipMemsetAsync(accum, 0, zero_bytes, stream);

  k1_row_stats<<<BATCH, 256, 0, stream>>>(logits, labels, rowmax, sumexp_t, accum + 0);
  k2_segsum  <<<BATCH, 256, 0, stream>>>(features, labels, seg_sum, counts);
  k3_centers <<<NCLS, 256, 0, stream>>>(centers, seg_sum, counts, centers_n);
  dim3 b4(32, 4), g4(NTILES, (NTILES + 3) / 4);
  k4_sim_wmma<<<g4, b4, 0, stream>>>(centers_n, sim);
  k5_contrast<<<BATCH, 256, 0, stream>>>(logits, labels, rowmax, sumexp_t, sim, accum + 1);
  k6_final   <<<1, 1, 0, stream>>>(accum, out);
}